// NbrAttn1_29532195127722
// MI455X (gfx1250) — compile-verified
//
#include <hip/hip_runtime.h>
#include <hip/hip_bf16.h>
#include <math.h>

// ---------------------------------------------------------------------------
// Problem constants (from reference): B=8, N=8, T=512, L=512, d=256, h=8, dk=32
// ---------------------------------------------------------------------------
#define BB   8
#define NN   8
#define TT   512
#define LL   512
#define DD   256
#define HH   8
#define DK   32
#define TSC  64
#define TSE  192
#define AUXC 16
#define AUXE 64
#define EDGC 8

typedef __attribute__((ext_vector_type(2))) float v2f;
typedef __attribute__((ext_vector_type(8))) float v8f;

// sinusoidal positional encoding value: pe[pos][j] for table of width d
__device__ __forceinline__ float pe_val(int pos, int j, int d) {
    int i = j >> 1;
    float div = __expf((-2.0f * (float)i) * (logf(10000.0f) / (float)d));
    float ang = (float)pos * div;
    return (j & 1) ? __cosf(ang) : __sinf(ang);
}

// ---------------------------------------------------------------------------
// Kernel 1: x_pe = x + PE(T,d);  q_in = layernorm(x_pe, ln_a, ln_b)  (ddof=1)
// ---------------------------------------------------------------------------
__global__ __launch_bounds__(DD) void k_prep(const float* __restrict__ x,
                                             const float* __restrict__ ln_a,
                                             const float* __restrict__ ln_b,
                                             float* __restrict__ xpe,
                                             float* __restrict__ qin) {
    int row = blockIdx.x;           // b*T + t
    int t   = row & (TT - 1);
    int j   = threadIdx.x;

    float v = x[(size_t)row * DD + j] + pe_val(t, j, DD);
    xpe[(size_t)row * DD + j] = v;

    __shared__ float red[DD];
    red[j] = v;
    __syncthreads();
    for (int s = DD / 2; s > 0; s >>= 1) {
        if (j < s) red[j] += red[j + s];
        __syncthreads();
    }
    float mu = red[0] * (1.0f / DD);
    __syncthreads();

    float dv = v - mu;
    red[j] = dv * dv;
    __syncthreads();
    for (int s = DD / 2; s > 0; s >>= 1) {
        if (j < s) red[j] += red[j + s];
        __syncthreads();
    }
    float sd = sqrtf(red[0] * (1.0f / (DD - 1)));   // ddof = 1
    qin[(size_t)row * DD + j] = ln_a[j] * dv / (sd + 1e-6f) + ln_b[j];
}

// ---------------------------------------------------------------------------
// Kernel 2: kv[b,n,l,:] = [ts_emb | aux_emb] + bias + PE(L,d)
// ---------------------------------------------------------------------------
__global__ __launch_bounds__(DD) void k_kv(const float* __restrict__ nbr_ts,
                                           const float* __restrict__ nbr_aux,
                                           const float* __restrict__ Wts,
                                           const float* __restrict__ bts,
                                           const float* __restrict__ Waux,
                                           const float* __restrict__ baux,
                                           float* __restrict__ kv) {
    int row = blockIdx.x;           // (b*N + n)*L + l
    int l   = row & (LL - 1);
    int bn  = row / LL;
    int e   = threadIdx.x;

    __shared__ float col_ts[TSC];
    __shared__ float col_aux[AUXC];
    if (e < TSC) {
        col_ts[e] = nbr_ts[((size_t)bn * TSC + e) * LL + l];
    } else if (e < TSC + AUXC) {
        col_aux[e - TSC] = nbr_aux[((size_t)bn * AUXC + (e - TSC)) * LL + l];
    }
    __syncthreads();

    float acc;
    if (e < TSE) {
        acc = bts[e];
        #pragma unroll 8
        for (int c = 0; c < TSC; ++c) acc += col_ts[c] * Wts[e * TSC + c];
    } else {
        int ee = e - TSE;
        acc = baux[ee];
        #pragma unroll
        for (int c = 0; c < AUXC; ++c) acc += col_aux[c] * Waux[ee * AUXC + c];
    }
    kv[(size_t)row * DD + e] = acc + pe_val(l, e, DD);
}

// ---------------------------------------------------------------------------
// Kernel 3: register-blocked, software-pipelined fp32 WMMA GEMM
//   C[M,N] = A[M,K] @ W[N,K]^T + bias
// One wave per 32x64 output block: 2 M-tiles x 4 N-tiles = 8 accumulators.
// Operand fragments are double-buffered: chunk k+1's 6 loads issue before
// chunk k's 8 WMMAs execute, so waits are on in-flight data, not full drains.
// VGPR layouts per CDNA5 ISA 7.12.2:
//   A 16x4 : lane(m=lane%16): vgpr0=A[m][2*half], vgpr1=A[m][2*half+1]
//   B 4x16 : lane(n=lane%16): vgpr0=B[2*half][n], vgpr1=B[2*half+1][n]
//   C 16x16: vgpr r: lanes0-15 -> C[r][n], lanes16-31 -> C[r+8][n]
// ---------------------------------------------------------------------------
__global__ __launch_bounds__(32) void k_gemm32x64(const float* __restrict__ A,
                                                  const float* __restrict__ W,
                                                  const float* __restrict__ bias,
                                                  float* __restrict__ C,
                                                  int M, int N, int K) {
    int tm   = blockIdx.x * 32;
    int tn   = blockIdx.y * 64;
    int lane = threadIdx.x;
    int half = lane >> 4;
    int idx  = lane & 15;

    const float* a0  = A + (size_t)(tm + idx) * K + 2 * half;
    const float* a1  = A + (size_t)(tm + 16 + idx) * K + 2 * half;
    const float* wr0 = W + (size_t)(tn +  0 + idx) * K + 2 * half;
    const float* wr1 = W + (size_t)(tn + 16 + idx) * K + 2 * half;
    const float* wr2 = W + (size_t)(tn + 32 + idx) * K + 2 * half;
    const float* wr3 = W + (size_t)(tn + 48 + idx) * K + 2 * half;

    v8f acc[2][4] = {};

    // prologue: load chunk 0 fragments
    v2f av0 = *(const v2f*)(a0);
    v2f av1 = *(const v2f*)(a1);
    v2f bv0 = *(const v2f*)(wr0);
    v2f bv1 = *(const v2f*)(wr1);
    v2f bv2 = *(const v2f*)(wr2);
    v2f bv3 = *(const v2f*)(wr3);

    for (int k0 = 0; k0 < K; k0 += 4) {
        // prefetch next chunk (clamped: last iteration re-reads in-bounds data)
        int kn = (k0 + 4 < K) ? (k0 + 4) : k0;
        v2f nav0 = *(const v2f*)(a0  + kn);
        v2f nav1 = *(const v2f*)(a1  + kn);
        v2f nbv0 = *(const v2f*)(wr0 + kn);
        v2f nbv1 = *(const v2f*)(wr1 + kn);
        v2f nbv2 = *(const v2f*)(wr2 + kn);
        v2f nbv3 = *(const v2f*)(wr3 + kn);

        acc[0][0] = __builtin_amdgcn_wmma_f32_16x16x4_f32(false, av0, false, bv0, (short)0, acc[0][0], false, false);
        acc[1][0] = __builtin_amdgcn_wmma_f32_16x16x4_f32(false, av1, false, bv0, (short)0, acc[1][0], false, false);
        acc[0][1] = __builtin_amdgcn_wmma_f32_16x16x4_f32(false, av0, false, bv1, (short)0, acc[0][1], false, false);
        acc[1][1] = __builtin_amdgcn_wmma_f32_16x16x4_f32(false, av1, false, bv1, (short)0, acc[1][1], false, false);
        acc[0][2] = __builtin_amdgcn_wmma_f32_16x16x4_f32(false, av0, false, bv2, (short)0, acc[0][2], false, false);
        acc[1][2] = __builtin_amdgcn_wmma_f32_16x16x4_f32(false, av1, false, bv2, (short)0, acc[1][2], false, false);
        acc[0][3] = __builtin_amdgcn_wmma_f32_16x16x4_f32(false, av0, false, bv3, (short)0, acc[0][3], false, false);
        acc[1][3] = __builtin_amdgcn_wmma_f32_16x16x4_f32(false, av1, false, bv3, (short)0, acc[1][3], false, false);

        av0 = nav0; av1 = nav1;
        bv0 = nbv0; bv1 = nbv1; bv2 = nbv2; bv3 = nbv3;
    }

    #pragma unroll
    for (int u = 0; u < 4; ++u) {
        float bn = bias[tn + 16 * u + idx];
        #pragma unroll
        for (int mi = 0; mi < 2; ++mi) {
            #pragma unroll
            for (int r = 0; r < 8; ++r) {
                int m = tm + 16 * mi + r + 8 * half;
                C[(size_t)m * N + tn + 16 * u + idx] = acc[mi][u][r] + bn;
            }
        }
    }
}

// ---------------------------------------------------------------------------
// Kernel 4: fused attention per (b, n, h, 16-row t-tile).  128 threads (4 waves).
//   phase 1: S(16x512) = Q(16x32) @ K^T -> LDS.  Q frags hoisted (loaded once);
//            K frags double-buffered across the 8 l-tiles per wave.
//   phase 2: mask + softmax over L in LDS
//   phase 3: ctx(16x32) = P @ V.  All 4 waves: wave w -> dk-tile (w&1),
//            K-half (w>>1); partials combined through LDS.
// ---------------------------------------------------------------------------
__global__ __launch_bounds__(128) void k_attn(const float* __restrict__ q,
                                              const float* __restrict__ kbuf,
                                              const float* __restrict__ vbuf,
                                              const int*   __restrict__ amask,
                                              float* __restrict__ ctx) {
    int bid = blockIdx.x;
    int tt = bid & 31;  bid >>= 5;   // T/16 = 32 tiles
    int hh = bid & 7;   bid >>= 3;
    int nn = bid & 7;   bid >>= 3;
    int bb = bid;

    int tid  = threadIdx.x;
    int wave = tid >> 5;
    int lane = tid & 31;
    int half = lane >> 4;
    int idx  = lane & 15;

    __shared__ float S[16][LL];
    __shared__ float red[16][8];
    __shared__ float rowmax[16];
    __shared__ float rowsum[16];
    __shared__ float part[2][16][16];

    const float* qrow  = q + ((size_t)bb * TT + tt * 16 + idx) * DD + hh * DK + 2 * half;
    const float* kbase = kbuf + (size_t)(bb * NN + nn) * LL * DD + hh * DK;

    // ---- phase 1: scores -> LDS --------------------------------------------
    // Q fragments are loop-invariant across the 8 l-tiles: load once.
    v2f qf[DK / 4];
    #pragma unroll
    for (int kc = 0; kc < DK / 4; ++kc)
        qf[kc] = *(const v2f*)(qrow + 4 * kc);

    // K fragments double-buffered across l-tiles.
    v2f bf[DK / 4], nbf[DK / 4];
    {
        const float* krow = kbase + (size_t)(wave * 8 * 16 + idx) * DD + 2 * half;
        #pragma unroll
        for (int kc = 0; kc < DK / 4; ++kc)
            bf[kc] = *(const v2f*)(krow + 4 * kc);
    }
    for (int i = 0; i < 8; ++i) {
        int lt = wave * 8 + i;
        // prefetch next l-tile (clamped on the last iteration)
        int ltn = wave * 8 + ((i < 7) ? (i + 1) : i);
        const float* knext = kbase + (size_t)(ltn * 16 + idx) * DD + 2 * half;
        #pragma unroll
        for (int kc = 0; kc < DK / 4; ++kc)
            nbf[kc] = *(const v2f*)(knext + 4 * kc);

        v8f acc = {};
        #pragma unroll
        for (int kc = 0; kc < DK / 4; ++kc)
            acc = __builtin_amdgcn_wmma_f32_16x16x4_f32(false, qf[kc], false, bf[kc],
                                                        (short)0, acc, false, false);
        #pragma unroll
        for (int r = 0; r < 8; ++r)
            S[r + 8 * half][lt * 16 + idx] = acc[r] * 0.17677669529663687f; // 1/sqrt(32)

        #pragma unroll
        for (int kc = 0; kc < DK / 4; ++kc) bf[kc] = nbf[kc];
    }
    __syncthreads();

    // ---- phase 2: mask + softmax over L ------------------------------------
    {
        int row = tid & 15;
        int sub = tid >> 4;                 // 0..7, each covers 64 columns
        const int* mrow = amask + (size_t)(bb * NN + nn) * LL;

        float mx = -3.0e38f;
        for (int c = sub * 64; c < sub * 64 + 64; ++c) {
            float vS = (mrow[c] != 0) ? S[row][c] : -1.0e9f;
            S[row][c] = vS;
            mx = fmaxf(mx, vS);
        }
        red[row][sub] = mx;
        __syncthreads();
        if (sub == 0) {
            float m2 = red[row][0];
            #pragma unroll
            for (int i = 1; i < 8; ++i) m2 = fmaxf(m2, red[row][i]);
            rowmax[row] = m2;
        }
        __syncthreads();
        float m2 = rowmax[row];
        float sm = 0.0f;
        for (int c = sub * 64; c < sub * 64 + 64; ++c) {
            float e = __expf(S[row][c] - m2);
            S[row][c] = e;
            sm += e;
        }
        red[row][sub] = sm;
        __syncthreads();
        if (sub == 0) {
            float s2 = 0.0f;
            #pragma unroll
            for (int i = 0; i < 8; ++i) s2 += red[row][i];
            rowsum[row] = s2;
        }
        __syncthreads();
        float inv = 1.0f / rowsum[row];
        for (int c = sub * 64; c < sub * 64 + 64; ++c) S[row][c] *= inv;
    }
    __syncthreads();

    // ---- phase 3: ctx = P @ V, all 4 waves ---------------------------------
    {
        int ntile = wave & 1;               // dk-tile (cols 0-15 / 16-31)
        int kh    = wave >> 1;              // K-half  (l in [0,256) / [256,512))
        const float* vb = vbuf + (size_t)(bb * NN + nn) * LL * DD
                               + hh * DK + ntile * 16 + idx;
        v8f acc = {};
        for (int k0 = kh * 256; k0 < kh * 256 + 256; k0 += 4) {
            v2f a, b;
            b.x = vb[(size_t)(k0 + 2 * half) * DD];
            b.y = vb[(size_t)(k0 + 2 * half + 1) * DD];
            a.x = S[idx][k0 + 2 * half];
            a.y = S[idx][k0 + 2 * half + 1];
            acc = __builtin_amdgcn_wmma_f32_16x16x4_f32(false, a, false, b,
                                                        (short)0, acc, false, false);
        }
        if (kh == 1) {
            #pragma unroll
            for (int r = 0; r < 8; ++r)
                part[ntile][r + 8 * half][idx] = acc[r];
        }
        __syncthreads();
        if (kh == 0) {
            float* cb = ctx + ((size_t)(bb * NN + nn) * TT + tt * 16) * DD
                            + hh * DK + ntile * 16 + idx;
            #pragma unroll
            for (int r = 0; r < 8; ++r)
                cb[(size_t)(r + 8 * half) * DD] = acc[r] + part[ntile][r + 8 * half][idx];
        }
    }
}

// ---------------------------------------------------------------------------
// Kernel 5: out[b,t,:] = mean_n ( (x_pe[b,t,:] + attn[b,n,t,:]) * edge[b,n,t,:] )
// edge[b,n,t,j] = sum_c nbr_edge[b,n,c,t] * Wedge[j,c] + bedge[j]
// ---------------------------------------------------------------------------
__global__ __launch_bounds__(DD) void k_final(const float* __restrict__ xpe,
                                              const float* __restrict__ attn,
                                              const float* __restrict__ nbr_edge,
                                              const float* __restrict__ Wedge,
                                              const float* __restrict__ bedge,
                                              float* __restrict__ out) {
    int row = blockIdx.x;           // b*T + t
    int b   = row / TT;
    int t   = row & (TT - 1);
    int j   = threadIdx.x;

    __shared__ float ecol[NN][EDGC];
    if (j < NN * EDGC) {
        int n = j >> 3, c = j & 7;
        ecol[n][c] = nbr_edge[((size_t)(b * NN + n) * EDGC + c) * TT + t];
    }
    __syncthreads();

    float xv = xpe[(size_t)row * DD + j];
    float be = bedge[j];
    float w[EDGC];
    #pragma unroll
    for (int c = 0; c < EDGC; ++c) w[c] = Wedge[j * EDGC + c];

    float acc = 0.0f;
    #pragma unroll
    for (int n = 0; n < NN; ++n) {
        float e = be;
        #pragma unroll
        for (int c = 0; c < EDGC; ++c) e += ecol[n][c] * w[c];
        float a = attn[((size_t)(b * NN + n) * TT + t) * DD + j];
        acc += (xv + a) * e;
    }
    out[(size_t)row * DD + j] = acc * (1.0f / NN);
}

// ---------------------------------------------------------------------------
// Launcher
// ---------------------------------------------------------------------------
extern "C" void kernel_launch(void* const* d_in, const int* in_sizes, int n_in,
                              void* d_out, int out_size, void* d_ws, size_t ws_size,
                              hipStream_t stream) {
    const float* x        = (const float*)d_in[0];
    const float* nbr_ts   = (const float*)d_in[1];
    const float* nbr_aux  = (const float*)d_in[2];
    const float* nbr_edge = (const float*)d_in[3];
    const int*   amask    = (const int*)  d_in[4];
    const float* Wts      = (const float*)d_in[5];
    const float* bts      = (const float*)d_in[6];
    const float* Waux     = (const float*)d_in[7];
    const float* baux     = (const float*)d_in[8];
    const float* Wedge    = (const float*)d_in[9];
    const float* bedge    = (const float*)d_in[10];
    const float* ln_a     = (const float*)d_in[11];
    const float* ln_b     = (const float*)d_in[12];
    const float* Wq       = (const float*)d_in[13];
    const float* bq       = (const float*)d_in[14];
    const float* Wk       = (const float*)d_in[15];
    const float* bk       = (const float*)d_in[16];
    const float* Wv       = (const float*)d_in[17];
    const float* bv       = (const float*)d_in[18];
    const float* Wo       = (const float*)d_in[19];
    const float* bo       = (const float*)d_in[20];
    float* out = (float*)d_out;

    // workspace carve-up (floats).  Total live high-water: ~28.3M floats (108MB)
    float* ws   = (float*)d_ws;
    float* xpe  = ws;                      // 8*512*256          = 1,048,576
    float* qin  = ws + 1048576;            //                      1,048,576
    float* qb   = ws + 2097152;            //                      1,048,576
    float* kvb  = ws + 3145728;            // 8*8*512*256        = 8,388,608
    float* kb   = ws + 11534336;           //                      8,388,608
    float* vb   = ws + 19922944;           //                      8,388,608
    float* ctx  = kvb;                     // reuse: kv dead after k/v projections
    float* attn = kb;                      // reuse: k dead after attention

    const int Mq  = BB * TT;               // 4096
    const int Mkv = BB * NN * LL;          // 32768

    // 1. PE + layernorm
    k_prep<<<Mq, DD, 0, stream>>>(x, ln_a, ln_b, xpe, qin);
    // 2. neighbor embeddings -> kv
    k_kv<<<Mkv, DD, 0, stream>>>(nbr_ts, nbr_aux, Wts, bts, Waux, baux, kvb);
    // 3. projections (fp32 WMMA GEMMs, 32x64 register blocking, pipelined)
    dim3 gq(Mq / 32, DD / 64);
    k_gemm32x64<<<gq, 32, 0, stream>>>(qin, Wq, bq, qb, Mq, DD, DD);
    dim3 gkv(Mkv / 32, DD / 64);
    k_gemm32x64<<<gkv, 32, 0, stream>>>(kvb, Wk, bk, kb, Mkv, DD, DD);
    k_gemm32x64<<<gkv, 32, 0, stream>>>(kvb, Wv, bv, vb, Mkv, DD, DD);
    // 4. fused masked softmax attention (QK^T + softmax + PV), WMMA
    k_attn<<<BB * NN * HH * (TT / 16), 128, 0, stream>>>(qb, kb, vb, amask, ctx);
    // 5. output projection (WMMA)
    k_gemm32x64<<<gkv, 32, 0, stream>>>(ctx, Wo, bo, attn, Mkv, DD, DD);
    // 6. edge gate + residual + mean over neighbors
    k_final<<<Mq, DD, 0, stream>>>(xpe, attn, nbr_edge, Wedge, bedge, out);
}